// SparseAttentionGate_37374805410614
// MI455X (gfx1250) — compile-verified
//
#include <hip/hip_runtime.h>
#include <math.h>
#include <stdint.h>

#define NLAYERS 7

typedef float vf4 __attribute__((ext_vector_type(4)));
typedef float v2f __attribute__((ext_vector_type(2)));
typedef float v8f __attribute__((ext_vector_type(8)));

#if __has_builtin(__builtin_amdgcn_global_load_async_to_lds_b32)
#define HAVE_ASYNC_LDS 1
#else
#define HAVE_ASYNC_LDS 0
#endif

typedef __attribute__((address_space(1))) int* gint_p;
typedef __attribute__((address_space(3))) int* lint_p;

// ---------------------------------------------------------------------------
// Kernel 1: stream 256 MB of zeros with nontemporal b128 stores.
// Dominant cost: ~11 us at 23.3 TB/s.  Everything else hides behind it.
// ---------------------------------------------------------------------------
__global__ __launch_bounds__(256) void sag_zerofill(vf4* __restrict__ out, int n4) {
  int i = blockIdx.x * blockDim.x + threadIdx.x;
  int stride = gridDim.x * blockDim.x;
  vf4 z = {0.f, 0.f, 0.f, 0.f};
  for (; i < n4; i += stride)
    __builtin_nontemporal_store(z, out + i);
}

// ---------------------------------------------------------------------------
// Kernel 2: one workgroup per (b, qi).  Gather <=4 neighbor blocks to LDS
// (async-to-LDS, ASYNCcnt-tracked), run the 7-layer residual 1x3 conv as f32
// WMMA GEMMs (V_WMMA_F32_16X16X4_F32, K=24 in 6 chunks), gate, deterministic
// row sums, normalize, scatter.
//
// Layout tricks that remove all predication from the hot loop:
//  * activation rows are stride-10 with zero guards at both ends, so the
//    w+/-1 im2col read is always a valid LDS address (guards supply the 0s)
//  * weight matrix rows co=8..15 replicate rows 0..7, so lanes 16-31 of the
//    WMMA D tile duplicate lanes 0-15 exactly -> no exec-masked stores
// ---------------------------------------------------------------------------
__global__ __launch_bounds__(128) void sag_compute(const float* __restrict__ in,
                                                   const float* __restrict__ cw,
                                                   const float* __restrict__ cb,
                                                   const float* __restrict__ pa,
                                                   float* __restrict__ out) {
  __shared__ float s_w2[NLAYERS * 16 * 8 * 3]; // [l][co(16, replicated)][ci][t]
  __shared__ float s_b[NLAYERS * 8];           // [l][co]
  __shared__ float s_al[NLAYERS];              // prelu alpha
  __shared__ float s_xc[4 * 8 * 8 * 10];       // [n][c][h][guard,w0..w7,guard]
  __shared__ float s_rs[64];                   // row sums [c][h]
  __shared__ int   s_nb[4];                    // neighbor kj list

  const int tid  = threadIdx.x;
  const int lane = tid & 31;
  const int wave = tid >> 5;
  const int b    = blockIdx.x >> 7;
  const int qi   = blockIdx.x & 127;

  if (tid == 0) {
    int m = 0;
    #pragma unroll
    for (int d = -2; d <= 2; ++d) {
      if (d == 0) continue;
      int kj = qi + d;
      if (kj >= 0 && kj < 128) s_nb[m++] = kj;
    }
  }
  // neighbor count, uniform, branch-free
  const int nn = (qi >= 2) + (qi >= 1) + (qi <= 126) + (qi <= 125);

  // zero-init activations (establishes the zero guard columns)
  for (int i = tid; i < 4 * 512 + 4 * 128; i += 128) s_xc[i] = 0.f;
  // weights: replicate co rows into co+8 (so A rows 8..15 mirror 0..7)
  for (int i = tid; i < NLAYERS * 192; i += 128) {
    float v = cw[i];
    int l = i / 192, r = i - l * 192;
    s_w2[l * 384 + r]       = v;
    s_w2[l * 384 + 192 + r] = v;
  }
  for (int i = tid; i < NLAYERS * 8; i += 128) s_b[i] = cb[i];
  if (tid < NLAYERS) s_al[tid] = pa[tid];
  __syncthreads();

  // ---- gather the selected blocks: x[n][c][h][1+w] -----------------------
  for (int r = tid; r < nn * 64; r += 128) {
    int n = r >> 6, rem = r & 63, c = rem >> 3, h = rem & 7;
    int gi = ((b * 8 + c) << 20) + ((qi * 8 + h) << 10) + (s_nb[n] << 3);
    int li = n * 640 + c * 80 + h * 10 + 1;
#if HAVE_ASYNC_LDS
    gint_p g3 = (gint_p)(uintptr_t)(const void*)(in + gi);
    lint_p l3 = (lint_p)(uint32_t)(uintptr_t)(const void*)(s_xc + li);
    #pragma unroll
    for (int w = 0; w < 8; ++w)
      __builtin_amdgcn_global_load_async_to_lds_b32(g3 + w, l3 + w, 0, 0);
#else
    vf4 lo = *(const vf4*)(in + gi);
    vf4 hi = *(const vf4*)(in + gi + 4);
    #pragma unroll
    for (int k = 0; k < 4; ++k) { s_xc[li + k] = lo[k]; s_xc[li + 4 + k] = hi[k]; }
#endif
  }
#if HAVE_ASYNC_LDS
#if __has_builtin(__builtin_amdgcn_s_wait_asynccnt)
  __builtin_amdgcn_s_wait_asynccnt(0);
#else
  asm volatile("s_wait_asynccnt 0x0" ::: "memory");
#endif
#endif
  __syncthreads();

  // ---- conv stack, one tile (= 2 (n,h) rows x 8 w = 16 columns) per wave --
  for (int t = wave; t < nn * 4; t += 4) {
    const int col  = lane & 15;          // B/D column index (N)
    const int r    = 2 * t + (col >> 3); // (n,h) row for this column
    const int n    = r >> 3, h = r & 7, w = col & 7;
    const int co   = lane & 15;          // A row (M); rows 8..15 replicated
    const int khi  = (lane >> 4) << 1;   // K offset for upper lane half
    const int xoff = n * 640 + h * 10 + 1 + w; // + v*80 selects channel

    // per-lane K -> (ci,t), K ordered t-major: K = t*8 + ci.  All addresses
    // are unconditionally valid (guards supply zeros, weights replicated).
    int baddr[12], wofft[12];
    #pragma unroll
    for (int kc = 0; kc < 6; ++kc) {
      #pragma unroll
      for (int j = 0; j < 2; ++j) {
        int K  = kc * 4 + j + khi;
        int ci = K & 7, tt = K >> 3;
        baddr[kc * 2 + j] = n * 640 + ci * 80 + h * 10 + w + tt; // (w-1+tt)+1
        wofft[kc * 2 + j] = co * 24 + ci * 3 + tt;
      }
    }

    float X[8], X0r[8];  // activation / original, channel v, column = lane
    #pragma unroll
    for (int v = 0; v < 8; ++v) { X[v] = s_xc[xoff + v * 80]; X0r[v] = X[v]; }

    #pragma unroll
    for (int l = 0; l < NLAYERS; ++l) {
      float av[12], bv[12];
      #pragma unroll
      for (int id = 0; id < 12; ++id) {
        av[id] = s_w2[l * 384 + wofft[id]];
        bv[id] = s_xc[baddr[id]];
      }
      v8f acc = {0.f, 0.f, 0.f, 0.f, 0.f, 0.f, 0.f, 0.f};
      #pragma unroll
      for (int kc = 0; kc < 6; ++kc) {
        v2f a2 = {av[2 * kc], av[2 * kc + 1]};
        v2f b2 = {bv[2 * kc], bv[2 * kc + 1]};
        acc = __builtin_amdgcn_wmma_f32_16x16x4_f32(
            false, a2, false, b2, (short)0, acc, false, false);
      }
      float alpha = s_al[l];
      #pragma unroll
      for (int v = 0; v < 8; ++v) {
        float y = acc[v] + s_b[l * 8 + v];
        y = (y >= 0.f) ? y : alpha * y;
        X[v] += y;                       // residual
        s_xc[xoff + v * 80] = X[v];      // lanes 16-31 duplicate lanes 0-15
      }
    }

    // ---- gate: we = exp(relu(x0 - sigmoid(conv))) - 1, diagonal masked ---
    #pragma unroll
    for (int v = 0; v < 8; ++v) {
      float sg = 1.f / (1.f + expf(-X[v]));
      float sp = X0r[v] - sg;
      sp = (sp > 0.f) ? sp : 0.f;
      if (h == w) sp = 0.f;              // self mask within A x A block
      s_xc[xoff + v * 80] = expf(sp) - 1.f;
    }
  }
  __syncthreads();

  // ---- deterministic row sums: thread (c,h) sums its row in fixed order --
  if (tid < 64) {
    int c = tid >> 3, h = tid & 7;
    float s = 0.f;
    for (int n = 0; n < nn; ++n) {
      #pragma unroll
      for (int w = 0; w < 8; ++w) s += s_xc[n * 640 + c * 80 + h * 10 + 1 + w];
    }
    s_rs[tid] = s;
  }
  __syncthreads();

  // ---- normalize + scatter over the zero-filled output -------------------
  for (int r = tid; r < nn * 64; r += 128) {
    int n = r >> 6, rem = r & 63, c = rem >> 3, h = rem & 7;
    float inv = 1.f / (s_rs[c * 8 + h] + 1e-5f);
    int li = n * 640 + c * 80 + h * 10 + 1;
    int gi = ((b * 8 + c) << 20) + ((qi * 8 + h) << 10) + (s_nb[n] << 3);
    vf4 o0, o1;
    #pragma unroll
    for (int k = 0; k < 4; ++k) {
      o0[k] = s_xc[li + k] * inv;
      o1[k] = s_xc[li + 4 + k] * inv;
    }
    *(vf4*)(out + gi)     = o0;
    *(vf4*)(out + gi + 4) = o1;
  }
}

extern "C" void kernel_launch(void* const* d_in, const int* in_sizes, int n_in,
                              void* d_out, int out_size, void* d_ws, size_t ws_size,
                              hipStream_t stream) {
  (void)in_sizes; (void)n_in; (void)d_ws; (void)ws_size;
  const float* pw = (const float*)d_in[0];  // (64, 1024, 1024) f32
  const float* cw = (const float*)d_in[1];  // (7,8,8,1,3) f32
  const float* cb = (const float*)d_in[2];  // (7,8) f32
  const float* pa = (const float*)d_in[3];  // (7,) f32
  float* out = (float*)d_out;

  int n4 = out_size >> 2;  // 16,777,216 float4 stores
  sag_zerofill<<<8192, 256, 0, stream>>>((vf4*)out, n4);
  sag_compute<<<8 * 128, 128, 0, stream>>>(pw, cw, cb, pa, out);
}